// CrossFuse_68135361184293
// MI455X (gfx1250) — compile-verified
//
#include <hip/hip_runtime.h>
#include <hip/hip_bf16.h>

typedef __attribute__((ext_vector_type(2))) float v2f;
typedef __attribute__((ext_vector_type(8))) float v8f;

#define N_SPATIAL 1024
#define C_HALF    2048
#define CHANNEL   4096
#define RED       256
#define BATCH     8

// ---------------- reductions (wave32) ----------------
__device__ __forceinline__ float waveSum(float v) {
#pragma unroll
    for (int m = 16; m > 0; m >>= 1) v += __shfl_xor(v, m, 32);
    return v;
}
__device__ __forceinline__ float waveMax(float v) {
#pragma unroll
    for (int m = 16; m > 0; m >>= 1) v = fmaxf(v, __shfl_xor(v, m, 32));
    return v;
}
// block of 256 threads = 8 waves
__device__ float blockReduce(float v, float* sm, bool isMax) {
    int lane = threadIdx.x & 31;
    int wid  = threadIdx.x >> 5;
    v = isMax ? waveMax(v) : waveSum(v);
    if (lane == 0) sm[wid] = v;
    __syncthreads();
    if (threadIdx.x == 0) {
        float r = sm[0];
#pragma unroll
        for (int i = 1; i < 8; i++) r = isMax ? fmaxf(r, sm[i]) : (r + sm[i]);
        sm[0] = r;
    }
    __syncthreads();
    float r = sm[0];
    __syncthreads();
    return r;
}

// ---------------- pass 1: per-row stats + analytic y ----------------
// row = b*2048 + c.  Stats for e1 and e2 rows of 1024 elements.
__global__ void __launch_bounds__(256)
stats_kernel(const float* __restrict__ emb1, const float* __restrict__ emb2,
             float* __restrict__ A1, float* __restrict__ M1,
             float* __restrict__ A2, float* __restrict__ M2,
             float* __restrict__ Y) {
    __shared__ __align__(16) float r1[N_SPATIAL];
    __shared__ __align__(16) float r2[N_SPATIAL];
    __shared__ float sm[8];
    const int row = blockIdx.x;
    const int t = threadIdx.x;
    const float* p1 = emb1 + (size_t)row * N_SPATIAL;
    const float* p2 = emb2 + (size_t)row * N_SPATIAL;
    // stage both rows in LDS (float4 coalesced)
    ((float4*)r1)[t] = ((const float4*)p1)[t];
    ((float4*)r2)[t] = ((const float4*)p2)[t];
    __syncthreads();

    float lmax1 = -3.402823466e38f, lmax2 = -3.402823466e38f;
#pragma unroll
    for (int i = t; i < N_SPATIAL; i += 256) {
        lmax1 = fmaxf(lmax1, r1[i]);
        lmax2 = fmaxf(lmax2, r2[i]);
    }
    const float max1 = blockReduce(lmax1, sm, true);
    const float max2 = blockReduce(lmax2, sm, true);

    float s1 = 0.f, q1 = 0.f, e1s = 0.f, x1 = 0.f;
    float s2 = 0.f, q2 = 0.f, e2s = 0.f, x2 = 0.f;
#pragma unroll
    for (int i = t; i < N_SPATIAL; i += 256) {
        float a = r1[i], b = r2[i];
        float ea = __expf(a - max1), eb = __expf(b - max2);
        s1 += a; q1 += a * a; e1s += ea; x1 += a * ea;
        s2 += b; q2 += b * b; e2s += eb; x2 += b * eb;
    }
    s1  = blockReduce(s1,  sm, false);
    q1  = blockReduce(q1,  sm, false);
    e1s = blockReduce(e1s, sm, false);
    x1  = blockReduce(x1,  sm, false);
    s2  = blockReduce(s2,  sm, false);
    q2  = blockReduce(q2,  sm, false);
    e2s = blockReduce(e2s, sm, false);
    x2  = blockReduce(x2,  sm, false);

    if (t == 0) {
        const float invn = 1.0f / N_SPATIAL;
        const float mean1 = s1 * invn, mean2 = s2 * invn;
        const float a1 = mean2 / e1s;   // coeff on exp(e1 - max1)
        const float a2 = mean1 / e2s;   // coeff on exp(e2 - max2)
        A1[row] = a1; M1[row] = max1;
        A2[row] = a2; M2[row] = max2;
        const int b_ = row / C_HALF, c_ = row % C_HALF;
        // y = mean_n(t): mean + sumsq/n^2 + A * sum(x*exp(x-max))/n
        Y[b_ * CHANNEL + c_]          = mean1 + q1 * invn * invn + a1 * x1 * invn;
        Y[b_ * CHANNEL + C_HALF + c_] = mean2 + q2 * invn * invn + a2 * x2 * invn;
    }
}

// zero y pad rows 8..15 (workspace is poisoned 0xAA by the harness)
__global__ void zero_pad_kernel(float* __restrict__ p) {
    p[blockIdx.x * 256 + threadIdx.x] = 0.0f;
}

// ---------------- pass 2a: hid = relu(Y @ w1^T), 16x256 via fp32 WMMA ----------------
// one wave per 16x16 output tile; 16 waves in one block. K = 4096, step 4.
__global__ void __launch_bounds__(512)
gemm1_kernel(const float* __restrict__ Y, const float* __restrict__ w1,
             float* __restrict__ HID) {
    const int wave  = threadIdx.x >> 5;   // 0..15 -> nbase
    const int lane  = threadIdx.x & 31;
    const int nbase = wave * 16;
    const int ml    = lane & 15;          // A row / B col within tile
    const int kk    = (lane >> 4) * 2;    // K-pair select (K=0,1 vs K=2,3)
    const float* ap = Y  + (size_t)ml * CHANNEL + kk;
    const float* bp = w1 + (size_t)(nbase + ml) * CHANNEL + kk;  // w1^T[k][n] = w1[n*4096+k]
    v8f acc = {};
#pragma unroll 4
    for (int k = 0; k < CHANNEL; k += 4) {
        v2f a = *(const v2f*)(ap + k);
        v2f b = *(const v2f*)(bp + k);
        acc = __builtin_amdgcn_wmma_f32_16x16x4_f32(
            false, a, false, b, (short)0, acc, false, false);
    }
    const int mrow = (lane >> 4) * 8;     // lanes 0-15: M=r, lanes 16-31: M=r+8
#pragma unroll
    for (int r = 0; r < 8; r++) {
        HID[(mrow + r) * RED + nbase + ml] = fmaxf(acc[r], 0.0f);
    }
}

// ---------------- pass 2b: mask = sigmoid(HID @ w2^T), 8x4096 via fp32 WMMA ----------------
// 256 tiles of 16x16; 32 blocks x 8 waves. K = 256.
__global__ void __launch_bounds__(256)
gemm2_kernel(const float* __restrict__ HID, const float* __restrict__ w2,
             float* __restrict__ MASK) {
    const int wave  = threadIdx.x >> 5;
    const int tile  = blockIdx.x * 8 + wave;  // 0..255
    const int lane  = threadIdx.x & 31;
    const int nbase = tile * 16;
    const int ml    = lane & 15;
    const int kk    = (lane >> 4) * 2;
    const float* ap = HID + (size_t)ml * RED + kk;
    const float* bp = w2  + (size_t)(nbase + ml) * RED + kk;  // w2^T[k][n] = w2[n*256+k]
    v8f acc = {};
#pragma unroll 8
    for (int k = 0; k < RED; k += 4) {
        v2f a = *(const v2f*)(ap + k);
        v2f b = *(const v2f*)(bp + k);
        acc = __builtin_amdgcn_wmma_f32_16x16x4_f32(
            false, a, false, b, (short)0, acc, false, false);
    }
    // only batch rows 0..7 are real -> elements r=0..7 on lanes 0..15
    if (lane < 16) {
#pragma unroll
        for (int r = 0; r < 8; r++) {
            float x = acc[r];
            MASK[r * CHANNEL + nbase + ml] = 1.0f / (1.0f + __expf(-x));
        }
    }
}

// ---------------- pass 3: fused elementwise output ----------------
// block per (b,ch) row; t = e*(1 + e/n + A*exp(e-max)); out = t*(1+mask)
__global__ void __launch_bounds__(256)
fused_out_kernel(const float* __restrict__ emb1, const float* __restrict__ emb2,
                 const float* __restrict__ A1, const float* __restrict__ M1,
                 const float* __restrict__ A2, const float* __restrict__ M2,
                 const float* __restrict__ MASK, float* __restrict__ out) {
    const int rc = blockIdx.x;            // b*4096 + ch
    const int b  = rc >> 12;
    const int ch = rc & 4095;
    const float* src;
    float A, mx;
    if (ch < C_HALF) {
        const int r = b * C_HALF + ch;
        src = emb1 + (size_t)r * N_SPATIAL;
        A = A1[r]; mx = M1[r];
    } else {
        const int r = b * C_HALF + (ch - C_HALF);
        src = emb2 + (size_t)r * N_SPATIAL;
        A = A2[r]; mx = M2[r];
    }
    const float g = 1.0f + MASK[rc];
    const float invn = 1.0f / N_SPATIAL;
    float4 e = ((const float4*)src)[threadIdx.x];
    float4 o;
    o.x = e.x * (1.0f + e.x * invn + A * __expf(e.x - mx)) * g;
    o.y = e.y * (1.0f + e.y * invn + A * __expf(e.y - mx)) * g;
    o.z = e.z * (1.0f + e.z * invn + A * __expf(e.z - mx)) * g;
    o.w = e.w * (1.0f + e.w * invn + A * __expf(e.w - mx)) * g;
    ((float4*)(out + (size_t)rc * N_SPATIAL))[threadIdx.x] = o;
}

extern "C" void kernel_launch(void* const* d_in, const int* in_sizes, int n_in,
                              void* d_out, int out_size, void* d_ws, size_t ws_size,
                              hipStream_t stream) {
    const float* emb1 = (const float*)d_in[0];  // (8,2048,32,32)
    const float* emb2 = (const float*)d_in[1];  // (8,2048,32,32)
    const float* w1   = (const float*)d_in[2];  // (256,4096)
    const float* w2   = (const float*)d_in[3];  // (4096,256)
    float* out = (float*)d_out;                 // (8,4096,32,32)

    const int NROWS = BATCH * C_HALF;           // 16384
    float* ws   = (float*)d_ws;
    float* A1   = ws;                           // 16384
    float* M1   = A1 + NROWS;                   // 16384
    float* A2   = M1 + NROWS;                   // 16384
    float* M2   = A2 + NROWS;                   // 16384
    float* Y    = M2 + NROWS;                   // 16*4096 (rows 8..15 zero pad)
    float* HID  = Y + 16 * CHANNEL;             // 16*256
    float* MASK = HID + 16 * RED;               // 8*4096

    // zero the 8 padded Y rows (8*4096 floats = 128 blocks * 256 threads)
    zero_pad_kernel<<<128, 256, 0, stream>>>(Y + 8 * CHANNEL);

    stats_kernel<<<NROWS, 256, 0, stream>>>(emb1, emb2, A1, M1, A2, M2, Y);

    gemm1_kernel<<<1, 512, 0, stream>>>(Y, w1, HID);
    gemm2_kernel<<<32, 256, 0, stream>>>(HID, w2, MASK);

    fused_out_kernel<<<BATCH * CHANNEL, 256, 0, stream>>>(
        emb1, emb2, A1, M1, A2, M2, MASK, out);
}